// NS_Flashed_TotalSimRetina_26448408609545
// MI455X (gfx1250) — compile-verified
//
#include <hip/hip_runtime.h>
#include <hip/hip_bf16.h>
#include <math.h>

// Problem constants (from reference)
#define BATCH    32
#define NCELLS   600
#define HW       4096     // 64*64, K dim of stimulus GEMM
#define NBF      100
#define MC       20
#define NB       500
#define NB_INIT  100
#define NSTEPS   400      // NB - NB_INIT
#define LCONV    401      // NB - NBF + 1
#define RING     101      // LDS ring depth (101 -> write slot never read same step)
#define LSTR     101      // LDS ring-row stride in floats (odd -> conflict-free banks)

#define OUT_SPIKES_ELEMS ((size_t)BATCH * NCELLS * NB)            // 9,600,000
#define LDS_BYTES        ((size_t)NCELLS * LSTR * sizeof(float))  // 242,400 B (CDNA5 320KB LDS)

typedef float v2f __attribute__((ext_vector_type(2)));
typedef float v8f __attribute__((ext_vector_type(8)));

typedef __attribute__((address_space(1))) int* gint_p;   // global
typedef __attribute__((address_space(3))) int* lint_p;   // LDS

#if defined(__HIP_DEVICE_COMPILE__) && \
    __has_builtin(__builtin_amdgcn_global_load_async_to_lds_b32) && \
    __has_builtin(__builtin_amdgcn_s_wait_asynccnt)
#define HAS_ASYNC_LDS 1
#else
#define HAS_ASYNC_LDS 0
#endif

// ---------------------------------------------------------------------------
// Kernel 1: stim_time_conv[c, t] = sum_f stim_time[t+f] * tc[c, f]   (600x401)
// ---------------------------------------------------------------------------
__global__ void timeconv_kernel(const float* __restrict__ stim_time,  // (500)
                                const float* __restrict__ tc,         // (600,100)
                                float* __restrict__ stc) {            // (600,401)
  int idx = blockIdx.x * blockDim.x + threadIdx.x;
  if (idx >= NCELLS * LCONV) return;
  int c = idx / LCONV;
  int t = idx - c * LCONV;
  const float* fl = tc + (size_t)c * NBF;
  float acc = 0.f;
#pragma unroll 4
  for (int f = 0; f < NBF; ++f) acc = fmaf(stim_time[t + f], fl[f], acc);
  stc[idx] = acc;
}

// ---------------------------------------------------------------------------
// Kernel 2: stim_applied = stim_flat (32x4096) @ spat_flat^T (4096x600)
// f32 GEMM -> V_WMMA_F32_16X16X4_F32, one wave per 16x16 tile.
// Fast path for full tiles (no column masking in the hot loop).
// ---------------------------------------------------------------------------
__global__ void __launch_bounds__(32)
stim_gemm_wmma(const float* __restrict__ stim,   // (32, 4096)
               const float* __restrict__ spat,   // (600, 4096)
               float* __restrict__ sa) {         // (32, 600)
  const int lane = threadIdx.x;          // one wave32 per block
  const int half = lane >> 4;
  const int l15  = lane & 15;
  const int m0 = blockIdx.y * 16;        // batch tile: 0 or 16
  const int n0 = blockIdx.x * 16;        // cell tile: 0..592
  const int colc  = n0 + l15;
  const bool colok = colc < NCELLS;

  const float* arow = stim + (size_t)(m0 + l15) * HW;
  const float* brow = spat + (size_t)(colok ? colc : 0) * HW;  // safe addr

  v8f acc = {};
  if (n0 + 16 <= NCELLS) {               // full tile: no masking needed
    for (int k = 0; k < HW; k += 4) {
      const int kk = k + 2 * half;       // even -> 8B aligned
      v2f a  = *(const v2f*)(arow + kk);
      v2f bv = *(const v2f*)(brow + kk);
      acc = __builtin_amdgcn_wmma_f32_16x16x4_f32(
          false, a, false, bv, (short)0, acc, false, false);
    }
  } else {                               // edge tile: zero invalid columns
    for (int k = 0; k < HW; k += 4) {
      const int kk = k + 2 * half;
      v2f a  = *(const v2f*)(arow + kk);
      v2f bv = *(const v2f*)(brow + kk);
      if (!colok) { bv.x = 0.f; bv.y = 0.f; }
      acc = __builtin_amdgcn_wmma_f32_16x16x4_f32(
          false, a, false, bv, (short)0, acc, false, false);
    }
  }
  if (colok) {
#pragma unroll
    for (int v = 0; v < 8; ++v)          // D: M = v + 8*half, N = l15
      sa[(size_t)(m0 + v + 8 * half) * NCELLS + colc] = acc[v];
  }
}

// ---------------------------------------------------------------------------
// ring_dot: dot(window_row_over_time, filter[0..99]) with ring of depth 101.
// slot(t+f) = head+f for f < n1, else f - n1, where n1 = min(100, 101-head).
// Filter rows are 400B (16B aligned) -> float4 loads (global_load_b128).
// Window reads stay scalar LDS (alignment varies with head).
// ---------------------------------------------------------------------------
__device__ __forceinline__ float ring_dot(const float* __restrict__ row,
                                          const float* __restrict__ fl,
                                          int head, int n1, float acc) {
  // segment 1: f in [0, n1), slot = head + f   (f starts 16B-aligned in fl)
  int f = 0;
  const int n1c = n1 & ~3;
  for (; f < n1c; f += 4) {
    const float4 fv = *(const float4*)(fl + f);
    const float* rp = row + head + f;
    acc = fmaf(rp[0], fv.x, acc);
    acc = fmaf(rp[1], fv.y, acc);
    acc = fmaf(rp[2], fv.z, acc);
    acc = fmaf(rp[3], fv.w, acc);
  }
  for (; f < n1; ++f) acc = fmaf(row[head + f], fl[f], acc);
  // segment 2: f in [n1, 100), slot = f - n1
  int f2 = n1;
  const int a4 = (n1 + 3) & ~3;
  for (; f2 < a4 && f2 < NBF; ++f2) acc = fmaf(row[f2 - n1], fl[f2], acc);
  for (; f2 < NBF; f2 += 4) {
    const float4 fv = *(const float4*)(fl + f2);
    const float* rp = row + (f2 - n1);
    acc = fmaf(rp[0], fv.x, acc);
    acc = fmaf(rp[1], fv.y, acc);
    acc = fmaf(rp[2], fv.z, acc);
    acc = fmaf(rp[3], fv.w, acc);
  }
  return acc;
}

// ---------------------------------------------------------------------------
// Kernel 3: 400-step GLM recurrence. One workgroup per batch (batches are
// independent). Full 600x101 spike ring lives in LDS (237 KB -- only possible
// with CDNA5's 320KB WGP LDS). Ring depth 101 means the slot written at step t
// (time t+100) is never read during step t -> single s_barrier per step.
// ---------------------------------------------------------------------------
__global__ void __launch_bounds__(640, 1)
recurrence_kernel(const float* __restrict__ init_spikes, // (B, C, 100)
                  const float* __restrict__ ff,          // (C, 100)
                  const float* __restrict__ cf,          // (C, 20, 100)
                  const float* __restrict__ bias,        // (C, 1)
                  const int*   __restrict__ sel,         // (C, 20)
                  const float* __restrict__ stc,         // (C, 401)  ws
                  const float* __restrict__ sa,          // (B, C)    ws
                  float* __restrict__ out_spikes,        // (B, C, 500)
                  float* __restrict__ out_gensig) {      // (B, C, 400)
  extern __shared__ float W[];                           // [NCELLS][LSTR] ring
  const int b   = blockIdx.x;
  const int tid = threadIdx.x;
  const int c   = tid;
  const bool active = (c < NCELLS);

  // Copy initial spikes to output (float4, rows 400B/2000B -> 16B aligned).
  for (int i4 = tid; i4 < NCELLS * NBF / 4; i4 += blockDim.x) {
    const int e  = i4 << 2;
    const int cc = e / NBF;
    const int f  = e - cc * NBF;
    const float4 v = *(const float4*)(init_spikes + ((size_t)b * NCELLS + cc) * NB_INIT + f);
    *(float4*)(out_spikes + ((size_t)b * NCELLS + cc) * NB + f) = v;
#if !HAS_ASYNC_LDS
    float* wp = &W[cc * LSTR + f];       // times 0..99 -> slots 0..99
    wp[0] = v.x; wp[1] = v.y; wp[2] = v.z; wp[3] = v.w;
#endif
  }
#if HAS_ASYNC_LDS
  // Fill the LDS ring via the CDNA5 async global->LDS DMA path (ASYNCcnt).
  for (int i = tid; i < NCELLS * NBF; i += blockDim.x) {
    const int cc = i / NBF;
    const int f  = i - cc * NBF;
    __builtin_amdgcn_global_load_async_to_lds_b32(
        (gint_p)(init_spikes + ((size_t)b * NCELLS + cc) * NB_INIT + f),
        (lint_p)&W[cc * LSTR + f],
        0, 0);
  }
  __builtin_amdgcn_s_wait_asynccnt(0);
#endif

  int   selreg[MC];
  float sa_bc = 0.f, bias_c = 0.f;
  if (active) {
#pragma unroll
    for (int m = 0; m < MC; ++m) selreg[m] = sel[c * MC + m];
    sa_bc  = sa[(size_t)b * NCELLS + c];
    bias_c = bias[c];
  }
  __syncthreads();

  const float* ffc = ff + (size_t)c * NBF;
  const float* cfc = cf + (size_t)c * MC * NBF;
  float* outg = out_gensig + ((size_t)b * NCELLS + c) * NSTEPS;
  float* outs = out_spikes + ((size_t)b * NCELLS + c) * NB + NBF;

  int head = 0;                                    // t % 101
  for (int t = 0; t < NSTEPS; ++t) {
    const int n1 = (101 - head < NBF) ? (101 - head) : NBF;
    if (active) {
      float acc = ring_dot(&W[c * LSTR], ffc, head, n1, 0.f);
      for (int m = 0; m < MC; ++m)
        acc = ring_dot(&W[selreg[m] * LSTR], cfc + m * NBF, head, n1, acc);
      const float g = acc + fmaf(sa_bc, stc[(size_t)c * LCONV + t], bias_c);
      const float spike = 1.f / (1.f + __expf(-g));
      outg[t] = g;
      outs[t] = spike;
      // time t+100 -> slot (head+100)%101 == head-1 (or 100): not read this step
      const int wslot = (head == 0) ? (RING - 1) : (head - 1);
      W[c * LSTR + wslot] = spike;
    }
    __syncthreads();                     // new spikes visible before next step
    head = (head == RING - 1) ? 0 : head + 1;
  }
}

// ---------------------------------------------------------------------------
extern "C" void kernel_launch(void* const* d_in, const int* in_sizes, int n_in,
                              void* d_out, int out_size, void* d_ws, size_t ws_size,
                              hipStream_t stream) {
  (void)in_sizes; (void)n_in; (void)out_size; (void)ws_size;
  const float* stim_frame  = (const float*)d_in[0];  // (32, 64, 64)
  const float* init_spikes = (const float*)d_in[1];  // (32, 600, 100)
  const float* spat        = (const float*)d_in[2];  // (600, 64, 64)
  const float* tc          = (const float*)d_in[3];  // (600, 100)
  const float* ff          = (const float*)d_in[4];  // (600, 100)
  const float* cf          = (const float*)d_in[5];  // (600, 20, 100)
  const float* bias        = (const float*)d_in[6];  // (600, 1)
  const int*   sel         = (const int*)d_in[7];    // (600, 20)
  const float* stim_time   = (const float*)d_in[8];  // (500)

  float* out_spikes = (float*)d_out;                       // (32,600,500)
  float* out_gensig = out_spikes + OUT_SPIKES_ELEMS;       // (32,600,400)

  float* stc = (float*)d_ws;                 // (600, 401)
  float* sa  = stc + (size_t)NCELLS * LCONV; // (32, 600)

  {
    const int total = NCELLS * LCONV;
    timeconv_kernel<<<(total + 255) / 256, 256, 0, stream>>>(stim_time, tc, stc);
  }
  {
    dim3 grid((NCELLS + 15) / 16, BATCH / 16);   // 38 x 2 tiles
    stim_gemm_wmma<<<grid, 32, 0, stream>>>(stim_frame, spat, sa);
  }
  recurrence_kernel<<<BATCH, 640, LDS_BYTES, stream>>>(
      init_spikes, ff, cf, bias, sel, stc, sa, out_spikes, out_gensig);
}